// ImgWordAtt_24240795418947
// MI455X (gfx1250) — compile-verified
//
#include <hip/hip_runtime.h>
#include <hip/hip_bf16.h>

typedef float v2f __attribute__((ext_vector_type(2)));
typedef float v8f __attribute__((ext_vector_type(8)));

#define EBD   512
#define ATT_H 512
#define NREG  196
#define NTOK  1024

// ---------------------------------------------------------------------------
// fast tanh: prefer the CDNA5 TRANS-pipe tanh; fall back to exp2+rcp (also
// TRANS ops) which saturates correctly at +/-1 for large |x|.
// ---------------------------------------------------------------------------
__device__ __forceinline__ float fast_tanh(float x) {
#if __has_builtin(__builtin_amdgcn_tanhf)
    return __builtin_amdgcn_tanhf(x);
#elif __has_builtin(__builtin_amdgcn_tanh_f32)
    return __builtin_amdgcn_tanh_f32(x);
#else
    // tanh(x) = 1 - 2/(1 + 2^(2x*log2e))
    float e = __builtin_amdgcn_exp2f(x * 2.885390081777927f);
    return 1.0f - 2.0f * __builtin_amdgcn_rcpf(e + 1.0f);
#endif
}

// ---------------------------------------------------------------------------
// K0: embedding gather. Writes E into workspace AND into out[:, 512:1024].
// ---------------------------------------------------------------------------
__global__ void gather_emb_kernel(const int* __restrict__ q,
                                  const float* __restrict__ table,
                                  float* __restrict__ E,
                                  float* __restrict__ out) {
    const int t = blockIdx.x;
    const float* src = table + (size_t)q[t] * EBD;
    for (int h = threadIdx.x; h < EBD; h += blockDim.x) {
        float v = src[h];
        E[(size_t)t * EBD + h] = v;
        out[(size_t)t * (2 * EBD) + EBD + h] = v;
    }
}

// ---------------------------------------------------------------------------
// Generic fp32 WMMA GEMM using V_WMMA_F32_16X16X4_F32 (wave32).
//   C[m,n] = scale * ( act( sum_k A[m,k]*B(k,n) + bias[n] ) )
//   BT=true : B stored [N,K] (C = A @ B^T)  -> float2 k-contiguous loads
//   BT=false: B stored [K,N] (C = A @ B)
//
// Each wave computes one 16x32 tile (1 A-fragment x 2 B-fragments): the A
// fragment is reused by two WMMAs per k-step. The k-loop is software-
// pipelined: loads for step k+4 are issued into fresh registers before the
// WMMAs of step k, so the hardware only needs a partial s_wait_loadcnt
// (previous step's loads) instead of a full stall before every WMMA.
//
// A-matrix per-lane layout (16x4 f32): row = lane&15, k = kbase..kbase+1,
// kbase = (lane>>4)*2.  C/D: vgpr v -> M = v + 8*(lane>>4), N = lane&15.
// ---------------------------------------------------------------------------
template <bool BT, bool RELU, bool HAS_BIAS>
__global__ __launch_bounds__(128) void wmma_gemm_kernel(
        const float* __restrict__ A, const float* __restrict__ Bm,
        const float* __restrict__ bias, float* __restrict__ C,
        int M, int N, int K, int ldc, float scale) {
    const int lane  = threadIdx.x & 31;
    const int wv    = threadIdx.x >> 5;          // 0..3
    const int mtile = blockIdx.x;
    const int ngrp  = blockIdx.y * 4 + wv;       // group of 2 n-tiles
    const int half  = lane >> 4;                 // 0: K 0..1, 1: K 2..3
    const int l16   = lane & 15;

    int arow = mtile * 16 + l16;
    if (arow >= M) arow = M - 1;                 // clamp pad rows (discarded at store)
    const int bcol0 = ngrp * 32 + l16;
    const int bcol1 = bcol0 + 16;

    const float* __restrict__ Arow = A + (size_t)arow * K + half * 2;

    auto loadA = [&](int kk) -> v2f {
        return *(const v2f*)(Arow + kk);
    };
    auto loadB = [&](int kk, int bcol) -> v2f {
        if (BT) {
            return *(const v2f*)(Bm + (size_t)bcol * K + kk + half * 2);
        } else {
            v2f b;
            const int kr = kk + half * 2;
            b.x = Bm[(size_t)kr * N + bcol];
            b.y = Bm[(size_t)(kr + 1) * N + bcol];
            return b;
        }
    };

    v8f acc0 = {};
    v8f acc1 = {};

    // pipeline prologue
    v2f a  = loadA(0);
    v2f b0 = loadB(0, bcol0);
    v2f b1 = loadB(0, bcol1);

#pragma unroll 4
    for (int k = 0; k < K - 4; k += 4) {
        // issue next step's loads first (fresh regs, stay in flight across WMMAs)
        v2f an  = loadA(k + 4);
        v2f b0n = loadB(k + 4, bcol0);
        v2f b1n = loadB(k + 4, bcol1);
        if (BT) __builtin_prefetch(Bm + (size_t)bcol0 * K + k + 36, 0, 1);
        acc0 = __builtin_amdgcn_wmma_f32_16x16x4_f32(false, a, false, b0,
                                                     (short)0, acc0, false, false);
        acc1 = __builtin_amdgcn_wmma_f32_16x16x4_f32(false, a, false, b1,
                                                     (short)0, acc1, false, false);
        a = an; b0 = b0n; b1 = b1n;
    }
    // pipeline epilogue
    acc0 = __builtin_amdgcn_wmma_f32_16x16x4_f32(false, a, false, b0,
                                                 (short)0, acc0, false, false);
    acc1 = __builtin_amdgcn_wmma_f32_16x16x4_f32(false, a, false, b1,
                                                 (short)0, acc1, false, false);

    const float bs0 = HAS_BIAS ? bias[bcol0] : 0.0f;
    const float bs1 = HAS_BIAS ? bias[bcol1] : 0.0f;
#pragma unroll
    for (int v = 0; v < 8; ++v) {
        const int m = mtile * 16 + half * 8 + v;
        if (m < M) {
            float v0 = acc0[v] + bs0;
            float v1 = acc1[v] + bs1;
            if (RELU) { v0 = fmaxf(v0, 0.0f); v1 = fmaxf(v1, 0.0f); }
            C[(size_t)m * ldc + bcol0] = v0 * scale;
            C[(size_t)m * ldc + bcol1] = v1 * scale;
        }
    }
}

// ---------------------------------------------------------------------------
// K4: additive-attention scores + softmax over regions.
//   s[t,r] = b_att + sum_h tanh(AI[r,h] + AW[t,h]) * w_att[h]
//   Wout[t,r] = softmax_r(s[t,r])
// One block (256 thr = 8 waves) per token. AW/w_att staged in LDS; lanes sweep
// h so AI reads are coalesced; wave32 shuffle reduction for the h-dot.
// ---------------------------------------------------------------------------
__global__ __launch_bounds__(256) void att_softmax_kernel(
        const float* __restrict__ AI, const float* __restrict__ AW,
        const float* __restrict__ w_att, const float* __restrict__ b_att,
        float* __restrict__ Wout) {
    const int t    = blockIdx.x;
    const int tid  = threadIdx.x;
    const int lane = tid & 31;
    const int wv   = tid >> 5;                   // 0..7

    __shared__ float aw_sh[ATT_H];
    __shared__ float wa_sh[ATT_H];
    __shared__ float sc_sh[200];
    __shared__ float red[256];

    for (int h = tid; h < ATT_H; h += 256) {
        aw_sh[h] = AW[(size_t)t * ATT_H + h];
        wa_sh[h] = w_att[h];
    }
    __syncthreads();

    const float batt = b_att[0];
    for (int r = wv; r < NREG; r += 8) {
        const float* ai = AI + (size_t)r * ATT_H;
        float s = 0.0f;
        for (int h = lane; h < ATT_H; h += 32)
            s += fast_tanh(ai[h] + aw_sh[h]) * wa_sh[h];
#pragma unroll
        for (int off = 16; off > 0; off >>= 1)
            s += __shfl_down(s, off, 32);
        if (lane == 0) sc_sh[r] = s + batt;
    }
    __syncthreads();

    // block-wide max over the 196 scores
    float v = (tid < NREG) ? sc_sh[tid] : -3.4e38f;
    red[tid] = v;
    __syncthreads();
    for (int s = 128; s > 0; s >>= 1) {
        if (tid < s) red[tid] = fmaxf(red[tid], red[tid + s]);
        __syncthreads();
    }
    const float mx = red[0];
    __syncthreads();

    const float e = (tid < NREG) ? __expf(v - mx) : 0.0f;
    red[tid] = e;
    __syncthreads();
    for (int s = 128; s > 0; s >>= 1) {
        if (tid < s) red[tid] += red[tid + s];
        __syncthreads();
    }
    const float inv = 1.0f / red[0];
    if (tid < NREG) Wout[(size_t)t * NREG + tid] = e * inv;
}

// ---------------------------------------------------------------------------
extern "C" void kernel_launch(void* const* d_in, const int* in_sizes, int n_in,
                              void* d_out, int out_size, void* d_ws, size_t ws_size,
                              hipStream_t stream) {
    const int*   question = (const int*)d_in[0];
    const float* img_feat = (const float*)d_in[1];  // [196, 512]
    const float* emb_tab  = (const float*)d_in[2];  // [VOCAB, 512]
    const float* W_img    = (const float*)d_in[3];  // [512, 512]
    const float* b_img    = (const float*)d_in[4];  // [512]
    const float* W_ai     = (const float*)d_in[5];  // [512, 512]
    const float* W_aw     = (const float*)d_in[6];  // [512, 512]
    const float* b_aw     = (const float*)d_in[7];  // [512]
    const float* w_att    = (const float*)d_in[8];  // [512]
    const float* b_att    = (const float*)d_in[9];  // [1]
    float*       out      = (float*)d_out;          // [1024, 1024]

    float* ws  = (float*)d_ws;
    float* E   = ws;                               // 1024*512
    float* AW  = E   + (size_t)NTOK * EBD;         // 1024*512
    float* IMG = AW  + (size_t)NTOK * ATT_H;       // 196*512
    float* AI  = IMG + (size_t)NREG * EBD;         // 196*512
    float* WAT = AI  + (size_t)NREG * ATT_H;       // 1024*196

    // K0: gather embeddings (also fills out[:, 512:1024])
    gather_emb_kernel<<<NTOK, 256, 0, stream>>>(question, emb_tab, E, out);

    // K1: IMG = relu(img_feat @ W_img^T + b_img)   [196, 512]
    wmma_gemm_kernel<true, true, true><<<dim3(13, 4), 128, 0, stream>>>(
        img_feat, W_img, b_img, IMG, NREG, EBD, 512, EBD, 1.0f);

    // K2: AI = IMG @ W_ai^T                        [196, 512]
    wmma_gemm_kernel<true, false, false><<<dim3(13, 4), 128, 0, stream>>>(
        IMG, W_ai, nullptr, AI, NREG, ATT_H, EBD, ATT_H, 1.0f);

    // K3: AW = E @ W_aw^T + b_aw                   [1024, 512]
    wmma_gemm_kernel<true, false, true><<<dim3(64, 4), 128, 0, stream>>>(
        E, W_aw, b_aw, AW, NTOK, ATT_H, EBD, ATT_H, 1.0f);

    // K4: scores + softmax -> WAT                  [1024, 196]
    att_softmax_kernel<<<NTOK, 256, 0, stream>>>(AI, AW, w_att, b_att, WAT);

    // K5: ctx = (WAT @ IMG) / 196 -> out[:, 0:512] (row stride 1024)
    wmma_gemm_kernel<false, false, false><<<dim3(64, 4), 128, 0, stream>>>(
        WAT, IMG, nullptr, out, NTOK, EBD, NREG, 2 * EBD, 1.0f / (float)NREG);
}